// TopKCrossEntropy_50474455662664
// MI455X (gfx1250) — compile-verified
//
#include <hip/hip_runtime.h>
#include <hip/hip_bf16.h>

// ---------------------------------------------------------------------------
// TopK cross-entropy for MI455X (gfx1250, wave32).
// Memory-bound: 134MB logits @ 23.3 TB/s ~ 6.5us floor. Exact top-k via
// 3-level radix select on float bits (13/10/9), counts-only histograms,
// then one compare-and-sum pass. CDNA5 paths: async global->LDS staging
// (ASYNCcnt) and v_wmma_f32_16x16x4_f32 wave reductions.
// ---------------------------------------------------------------------------

#define B_SZ 4
#define C_SZ 8
#define NVOX (64 * 128 * 128)   // 1,048,576 voxels per batch
#define NB1 8192                // bits[31:19]
#define NB2 1024                // bits[18:9]
#define NB3 512                 // bits[8:0]
#define NEG_INF_BITS 0xFF800000u

struct SelState {
  unsigned n_valid, k, count_above, bin1, bin2, thresh, pad0, pad1; // 32B
};

typedef __attribute__((ext_vector_type(2))) float v2f;
typedef __attribute__((ext_vector_type(8))) float v8f;

// ---- feature guards (device pass only; host pass always takes fallback) ----
#if defined(__HIP_DEVICE_COMPILE__) && __has_builtin(__builtin_amdgcn_global_load_async_to_lds_b32)
#define USE_ASYNC_LDS 1
#else
#define USE_ASYNC_LDS 0
#endif

#if defined(__HIP_DEVICE_COMPILE__) && __has_builtin(__builtin_amdgcn_wmma_f32_16x16x4_f32)
#define HAS_WMMA_K4 1
#else
#define HAS_WMMA_K4 0
#endif

#if USE_ASYNC_LDS
#if __has_builtin(__builtin_amdgcn_s_wait_asynccnt)
#define WAIT_ASYNC(n) __builtin_amdgcn_s_wait_asynccnt(n)
#else
#define WAIT_ASYNC(n) asm volatile("s_wait_asynccnt %0" ::"i"(n) : "memory")
#endif
// Builtin signature (per hipcc diagnostic): global-AS int* , LDS-AS int*.
typedef __attribute__((address_space(1))) int GInt;
typedef __attribute__((address_space(3))) int LInt;
__device__ __forceinline__ GInt* to_global(const void* p) {
  return (GInt*)(unsigned long long)p;   // int->ptr cast: explicit addrspace
}
__device__ __forceinline__ LInt* to_lds(void* p) {
  // generic LDS address: low 32 bits are the DS byte address (ISA 10.2)
  return (LInt*)(unsigned)(unsigned long long)p;
}
#endif

// ---- wave32 sum reduction ------------------------------------------------
// WMMA trick: A(16x4) with VGPR0 = lane partial, VGPR1 = 0, B = ones(4x16):
// D[m][n] = A[m][0]+A[m][2] = s_m + s_{m+16} replicated over n. Summing the
// 8 C/D VGPRs in a lane gives the half-sum; shfl_xor(16) completes it.
__device__ __forceinline__ float wave_sum32(float v) {
#if HAS_WMMA_K4
  v2f A; A.x = v;    A.y = 0.0f;
  v2f Bv; Bv.x = 1.0f; Bv.y = 1.0f;
  v8f Cv = {0.f, 0.f, 0.f, 0.f, 0.f, 0.f, 0.f, 0.f};
  v8f D = __builtin_amdgcn_wmma_f32_16x16x4_f32(false, A, false, Bv,
                                                (short)0, Cv, false, false);
  float s = D[0] + D[1] + D[2] + D[3] + D[4] + D[5] + D[6] + D[7];
  return s + __shfl_xor(s, 16, 32);
#else
  for (int m = 16; m >= 1; m >>= 1) v += __shfl_xor(v, m, 32);
  return v;
#endif
}

// ---- kernel 0: zero the histogram/state region of d_ws -------------------
__global__ __launch_bounds__(256) void zero_u32(unsigned* __restrict__ p, int n) {
  int i = blockIdx.x * 256 + threadIdx.x;
  if (i < n) p[i] = 0u;
}

// ---- kernel 1: CE per voxel + level-1 histogram (counts) -----------------
// Streams 8 channel planes; per voxel: logsumexp over C=8, nll = lse - x[t],
// invalid -> -inf in CE buffer. 8192-bin LDS histogram (bits>>19), privatized
// per block, flushed with skip-zero global atomics.
__global__ __launch_bounds__(256) void ce_hist1(
    const float* __restrict__ logits, const int* __restrict__ target,
    float* __restrict__ ce, unsigned* __restrict__ h1c) {
  const int b = blockIdx.y;
  const int tid = threadIdx.x;
  __shared__ unsigned sc[NB1];
  for (int i = tid; i < NB1; i += 256) sc[i] = 0u;
  __syncthreads();

  const float* lb = logits + (size_t)b * C_SZ * NVOX;
  const int* tb = target + (size_t)b * NVOX;
  float* cb = ce + (size_t)b * NVOX;
  const int stride = gridDim.x * 256;

#if USE_ASYNC_LDS
  __shared__ float stage[2][C_SZ * 256];
  {
    const int base0 = blockIdx.x * 256;
    const float* g = lb + base0 + tid;
#pragma unroll
    for (int c = 0; c < C_SZ; ++c)
      __builtin_amdgcn_global_load_async_to_lds_b32(
          to_global(g + (size_t)c * NVOX),
          to_lds(&stage[0][c * 256 + tid]), 0, 0);
  }
  int buf = 0;
  for (int base = blockIdx.x * 256; base < NVOX; base += stride) {
    const int nbase = base + stride;
    if (nbase < NVOX) {
      const float* g = lb + nbase + tid;
#pragma unroll
      for (int c = 0; c < C_SZ; ++c)
        __builtin_amdgcn_global_load_async_to_lds_b32(
            to_global(g + (size_t)c * NVOX),
            to_lds(&stage[buf ^ 1][c * 256 + tid]), 0, 0);
      WAIT_ASYNC(8);   // previous tile's 8 async loads complete (in-order)
    } else {
      WAIT_ASYNC(0);
    }
    asm volatile("" ::: "memory");   // compiler fence before LDS reads
    float x[C_SZ];
#pragma unroll
    for (int c = 0; c < C_SZ; ++c) x[c] = stage[buf][c * 256 + tid];
    buf ^= 1;
    const int idx = base + tid;
#else
  for (int base = blockIdx.x * 256; base < NVOX; base += stride) {
    const int idx = base + tid;
    float x[C_SZ];
#pragma unroll
    for (int c = 0; c < C_SZ; ++c) x[c] = lb[(size_t)c * NVOX + idx];
#endif
    const int tv = tb[idx];
    float m = x[0];
#pragma unroll
    for (int c = 1; c < C_SZ; ++c) m = fmaxf(m, x[c]);
    float sum = 0.0f;
#pragma unroll
    for (int c = 0; c < C_SZ; ++c) sum += __expf(x[c] - m);
    const float lse = m + __logf(sum);
    const int ti = tv & 7;
    const float s01 = (ti & 1) ? x[1] : x[0];
    const float s23 = (ti & 1) ? x[3] : x[2];
    const float s45 = (ti & 1) ? x[5] : x[4];
    const float s67 = (ti & 1) ? x[7] : x[6];
    const float q0 = (ti & 2) ? s23 : s01;
    const float q1 = (ti & 2) ? s67 : s45;
    const float xt = (ti & 4) ? q1 : q0;
    const float nll = fmaxf(lse - xt, 0.0f);   // CE >= 0 => bits monotonic
    const bool valid = (tv != 0);
    cb[idx] = valid ? nll : __uint_as_float(NEG_INF_BITS);
    if (valid) atomicAdd(&sc[__float_as_uint(nll) >> 19], 1u);
  }
  __syncthreads();
  unsigned* gc = h1c + (size_t)b * NB1;
  for (int i = tid; i < NB1; i += 256)
    if (sc[i]) atomicAdd(&gc[i], sc[i]);
}

// ---- kernels 3/5: refine histogram within the selected prefix ------------
__global__ __launch_bounds__(256) void hist_refine(
    const float* __restrict__ ce, const SelState* __restrict__ st,
    unsigned* __restrict__ hc, int stage) {
  const int b = blockIdx.y;
  const int tid = threadIdx.x;
  const int nbins = (stage == 0) ? NB2 : NB3;
  __shared__ unsigned sc[NB2];
  for (int i = tid; i < nbins; i += 256) sc[i] = 0u;
  const SelState S = st[b];
  __syncthreads();
  if (S.n_valid == 0u) return;   // uniform early-out

  const unsigned pref = (stage == 0) ? S.bin1 : ((S.bin1 << 10) | S.bin2);
  const int prefShift = (stage == 0) ? 19 : 9;
  const int binShift  = (stage == 0) ? 9 : 0;
  const unsigned binMask = (stage == 0) ? 0x3FFu : 0x1FFu;
  const float* cb = ce + (size_t)b * NVOX;
  const int stride = gridDim.x * 256;
  for (int idx = blockIdx.x * 256 + tid; idx < NVOX; idx += stride) {
    const unsigned bits = __float_as_uint(cb[idx]);
    if ((int)bits >= 0 && (bits >> prefShift) == pref)
      atomicAdd(&sc[(bits >> binShift) & binMask], 1u);
  }
  __syncthreads();
  unsigned* gc = hc + (size_t)b * nbins;
  for (int i = tid; i < nbins; i += 256)
    if (sc[i]) atomicAdd(&gc[i], sc[i]);
}

// ---- kernels 2/4/6: walk a histogram from the top to locate the k-th -----
__global__ __launch_bounds__(64) void select_k(
    const unsigned* __restrict__ hc, SelState* __restrict__ st,
    int nbins, int level) {
  const int b = blockIdx.x;
  const int tid = threadIdx.x;
  __shared__ unsigned segC[64];
  const unsigned* c = hc + (size_t)b * nbins;
  const int spt = nbins >> 6;
  unsigned tc = 0;
  for (int i = 0; i < spt; ++i) tc += c[tid * spt + i];
  segC[tid] = tc;
  __syncthreads();
  if (tid != 0) return;

  SelState S = st[b];
  unsigned kr;
  if (level == 0) {
    unsigned nv = 0;
    for (int i = 0; i < 64; ++i) nv += segC[i];
    S.n_valid = nv;
    int kk = (int)((float)nv * 0.2f);          // f32 mul + truncation: == ref
    if (kk < 1) kk = 1;
    unsigned k = (unsigned)kk;
    if (k > nv) k = nv;
    S.k = k;
    S.count_above = 0u;
    if (nv == 0u) { st[b] = S; return; }
    kr = k;
  } else {
    if (S.n_valid == 0u) return;
    kr = S.k - S.count_above;                   // remaining, >= 1
  }
  unsigned acc = 0;
  int seg = 63;
  while (seg > 0 && acc + segC[seg] < kr) { acc += segC[seg]; --seg; }
  const int lo = seg * spt;
  int bin = lo + spt - 1;
  while (bin > lo && acc + c[bin] < kr) { acc += c[bin]; --bin; }
  S.count_above += acc;                         // strictly-above count
  if (level == 0)      S.bin1 = (unsigned)bin;
  else if (level == 1) S.bin2 = (unsigned)bin;
  else                 S.thresh = (S.bin1 << 19) | (S.bin2 << 9) | (unsigned)bin;
  st[b] = S;
}

// ---- kernel 7: sum of all CE strictly greater than threshold -------------
__global__ __launch_bounds__(256) void final_sum(
    const float* __restrict__ ce, const SelState* __restrict__ st,
    float* __restrict__ sumAbove) {
  const int b = blockIdx.y;
  const SelState S = st[b];
  if (S.n_valid == 0u) return;
  const unsigned T = S.thresh;
  const float* cb = ce + (size_t)b * NVOX;
  float acc = 0.0f;
  const int stride = gridDim.x * 256;
  for (int idx = blockIdx.x * 256 + threadIdx.x; idx < NVOX; idx += stride) {
    const unsigned bits = __float_as_uint(cb[idx]);
    if ((int)bits >= 0 && bits > T) acc += __uint_as_float(bits);
  }
  const float wsum = wave_sum32(acc);           // WMMA-based wave reduction
  __shared__ float red[8];
  const int wid = threadIdx.x >> 5, lid = threadIdx.x & 31;
  if (lid == 0) red[wid] = wsum;
  __syncthreads();
  if (threadIdx.x == 0) {
    float t = 0.0f;
#pragma unroll
    for (int i = 0; i < 8; ++i) t += red[i];
    atomicAdd(&sumAbove[b], t);
  }
}

// ---- kernel 8: per-sample means (ties at threshold) + batch mean ---------
__global__ __launch_bounds__(32) void finalize(
    const SelState* __restrict__ st, const float* __restrict__ sumAbove,
    float* __restrict__ out) {
  const int lane = threadIdx.x;
  float v = 0.0f;
  if (lane < B_SZ) {
    const SelState S = st[lane];
    if (S.n_valid != 0u) {
      const float tvf = __uint_as_float(S.thresh);
      const unsigned tie = S.k - S.count_above;   // >= 1 copies of threshold
      v = (sumAbove[lane] + (float)tie * tvf) / (float)S.k;
    }
  }
  const float total = wave_sum32(v);              // WMMA-based wave reduction
  if (lane == 0) out[0] = total * (1.0f / (float)B_SZ);
}

// ---------------------------------------------------------------------------
extern "C" void kernel_launch(void* const* d_in, const int* in_sizes, int n_in,
                              void* d_out, int out_size, void* d_ws, size_t ws_size,
                              hipStream_t stream) {
  (void)in_sizes; (void)n_in; (void)out_size; (void)ws_size;
  const float* logits = (const float*)d_in[0];
  const int* target = (const int*)d_in[1];
  float* out = (float*)d_out;

  // workspace layout (all 4-byte aligned, contiguous):
  float* ce = (float*)d_ws;                                   // B*N f32 (16MB)
  unsigned* h1c = (unsigned*)(ce + (size_t)B_SZ * NVOX);      // B*8192
  unsigned* h2c = h1c + B_SZ * NB1;                           // B*1024
  unsigned* h3c = h2c + B_SZ * NB2;                           // B*512
  float* sumAbove = (float*)(h3c + B_SZ * NB3);               // B floats
  SelState* st = (SelState*)(sumAbove + B_SZ);                // B*32 bytes

  const int zeroN = B_SZ * (NB1 + NB2 + NB3) + B_SZ +
                    B_SZ * (int)(sizeof(SelState) / 4);
  zero_u32<<<(zeroN + 255) / 256, 256, 0, stream>>>(h1c, zeroN);

  dim3 gA(512, B_SZ);
  ce_hist1<<<gA, 256, 0, stream>>>(logits, target, ce, h1c);
  select_k<<<B_SZ, 64, 0, stream>>>(h1c, st, NB1, 0);
  dim3 gR(256, B_SZ);
  hist_refine<<<gR, 256, 0, stream>>>(ce, st, h2c, 0);
  select_k<<<B_SZ, 64, 0, stream>>>(h2c, st, NB2, 1);
  hist_refine<<<gR, 256, 0, stream>>>(ce, st, h3c, 1);
  select_k<<<B_SZ, 64, 0, stream>>>(h3c, st, NB3, 2);
  final_sum<<<gR, 256, 0, stream>>>(ce, st, sumAbove);
  finalize<<<1, 32, 0, stream>>>(st, sumAbove, out);
}